// Sparse_attention_8950711845760
// MI455X (gfx1250) — compile-verified
//
#include <hip/hip_runtime.h>
#include <hip/hip_bf16.h>
#include <cstdint>

#define TIME   1024
#define TPB    128          // 4 wave32s: 1 wave per SIMD32
#define NWAVES (TPB / 32)
#define VPT    8            // values per thread
#define EPSF   1e-7f

typedef __attribute__((ext_vector_type(4))) float f32x4;

// Branchless compare-exchange: max -> lower index (descending order).
#define CSWP(a, i, j)                                   \
    do {                                                \
        float _hi = fmaxf((a)[i], (a)[j]);              \
        float _lo = fminf((a)[i], (a)[j]);              \
        (a)[i] = _hi; (a)[j] = _lo;                     \
    } while (0)

// XOR-butterfly lane exchange via DS_SWIZZLE_B32 group-of-32 mode:
// offset imm = { bit15=0, xor_mask[14:10], or_mask[9:5]=0, and_mask[4:0]=0x1f }.
// One DS op, no VALU address math (vs __shfl_xor's mbcnt/bitop/cmp/bpermute).
template <int XOR>
__device__ __forceinline__ float swzx(float x) {
    int r = __builtin_amdgcn_ds_swizzle(__float_as_int(x), (XOR << 10) | 0x1f);
    return __int_as_float(r);
}

// Batcher odd-even mergesort, 8 inputs, 19 CE -> descending.
__device__ __forceinline__ void sort8(float (&v)[8]) {
    CSWP(v,0,1); CSWP(v,2,3); CSWP(v,4,5); CSWP(v,6,7);
    CSWP(v,0,2); CSWP(v,1,3); CSWP(v,4,6); CSWP(v,5,7);
    CSWP(v,1,2); CSWP(v,5,6);
    CSWP(v,0,4); CSWP(v,1,5); CSWP(v,2,6); CSWP(v,3,7);
    CSWP(v,2,4); CSWP(v,3,5);
    CSWP(v,1,2); CSWP(v,3,4); CSWP(v,5,6);
}

// Merge two sorted-descending 6-lists; t := top-6 of union, sorted descending.
// Half-cleaner (exact top-6 multiset; output is V-shaped bitonic), then a
// distance-3 half-clean (valid for any even-length bitonic sequence) which
// leaves top-3 >= bottom-3 elementwise with both halves bitonic, then two
// independent sort-3s. 6 max + 9 CE total.
__device__ __forceinline__ void merge6(float (&t)[6], const float (&o)[6]) {
    float m[6];
#pragma unroll
    for (int i = 0; i < 6; ++i) m[i] = fmaxf(t[i], o[5 - i]);
    CSWP(m,0,3); CSWP(m,1,4); CSWP(m,2,5);   // half-clean, distance 3
    CSWP(m,0,1); CSWP(m,1,2); CSWP(m,0,1);   // sort3 top
    CSWP(m,3,4); CSWP(m,4,5); CSWP(m,3,4);   // sort3 bottom
#pragma unroll
    for (int i = 0; i < 6; ++i) t[i] = m[i];
}

__global__ __launch_bounds__(TPB)
void sparse_attn_topk_norm(const float* __restrict__ src,
                           float* __restrict__ dst,
                           int rows) {
    const int row  = blockIdx.x;
    if (row >= rows) return;
    const int tid  = threadIdx.x;
    const int lane = tid & 31;
    const int wid  = tid >> 5;

    __shared__ __align__(16) float s_row[TIME];      // 4 KB row stage
    __shared__ float s_top[NWAVES * 8];              // per-wave top-6 (stride 8)
    __shared__ float s_red[NWAVES];
    __shared__ float s_bcast[2];                     // [0]=delta, [1]=denom

    const float* gsrc = src + (size_t)row * TIME + (size_t)tid * VPT;

    // ---- CDNA5 async bulk load: global -> LDS (ASYNCcnt path) ----
    // INST_OFFSET applies to both the LDS and global sides, so one address
    // pair covers both 16B chunks of this thread's 32B slice.
    unsigned lds_off = (unsigned)(uintptr_t)(void*)s_row + (unsigned)(tid * (VPT * 4));
    unsigned long long gaddr = (unsigned long long)(uintptr_t)gsrc;
    asm volatile("global_load_async_to_lds_b128 %0, %1, off"
                 :: "v"(lds_off), "v"(gaddr) : "memory");
    asm volatile("global_load_async_to_lds_b128 %0, %1, off offset:16"
                 :: "v"(lds_off), "v"(gaddr) : "memory");
    asm volatile("s_wait_asynccnt 0" ::: "memory");
    __syncthreads();

    // Pull this thread's 8 elements into registers (kept for both passes).
    f32x4 xa = *reinterpret_cast<const f32x4*>(&s_row[tid * VPT]);
    f32x4 xb = *reinterpret_cast<const f32x4*>(&s_row[tid * VPT + 4]);

    // ---- Pass 1: exact row top-6 (branchless networks) ----
    float v[8] = { xa.x, xa.y, xa.z, xa.w, xb.x, xb.y, xb.z, xb.w };
    sort8(v);

    float t[6];
#pragma unroll
    for (int k = 0; k < 6; ++k) t[k] = v[k];   // thread's top-6 (of 8), sorted

    // wave32 XOR-butterfly: merge sorted 6-lists across lanes via ds_swizzle
    {
        float o[6];
#pragma unroll
        for (int k = 0; k < 6; ++k) o[k] = swzx<16>(t[k]);
        merge6(t, o);
#pragma unroll
        for (int k = 0; k < 6; ++k) o[k] = swzx<8>(t[k]);
        merge6(t, o);
#pragma unroll
        for (int k = 0; k < 6; ++k) o[k] = swzx<4>(t[k]);
        merge6(t, o);
#pragma unroll
        for (int k = 0; k < 6; ++k) o[k] = swzx<2>(t[k]);
        merge6(t, o);
#pragma unroll
        for (int k = 0; k < 6; ++k) o[k] = swzx<1>(t[k]);
        merge6(t, o);
    }
    if (lane == 0) {
#pragma unroll
        for (int k = 0; k < 6; ++k) s_top[wid * 8 + k] = t[k];
    }
    __syncthreads();

    if (tid == 0) {
        float f[6];
#pragma unroll
        for (int k = 0; k < 6; ++k) f[k] = s_top[k];
        for (int w = 1; w < NWAVES; ++w) {
            float o[6];
#pragma unroll
            for (int k = 0; k < 6; ++k) o[k] = s_top[w * 8 + k];
            merge6(f, o);
        }
        s_bcast[0] = f[5] + EPSF;     // delta = 6th largest + eps
    }
    __syncthreads();
    const float delta = s_bcast[0];

    // ---- Pass 2: subtract+clamp, row sum, normalize ----
    float w0 = fmaxf(xa.x - delta, 0.0f);
    float w1 = fmaxf(xa.y - delta, 0.0f);
    float w2 = fmaxf(xa.z - delta, 0.0f);
    float w3 = fmaxf(xa.w - delta, 0.0f);
    float w4 = fmaxf(xb.x - delta, 0.0f);
    float w5 = fmaxf(xb.y - delta, 0.0f);
    float w6 = fmaxf(xb.z - delta, 0.0f);
    float w7 = fmaxf(xb.w - delta, 0.0f);

    float psum = ((w0 + w1) + (w2 + w3)) + ((w4 + w5) + (w6 + w7));
    psum += swzx<16>(psum);
    psum += swzx<8>(psum);
    psum += swzx<4>(psum);
    psum += swzx<2>(psum);
    psum += swzx<1>(psum);
    if (lane == 0) s_red[wid] = psum;
    __syncthreads();
    if (tid == 0) {
        float s = s_red[0];
        for (int w = 1; w < NWAVES; ++w) s += s_red[w];
        s_bcast[1] = s + EPSF;
    }
    __syncthreads();
    const float inv = 1.0f / s_bcast[1];

    f32x4 oa, ob;
    oa.x = w0 * inv; oa.y = w1 * inv; oa.z = w2 * inv; oa.w = w3 * inv;
    ob.x = w4 * inv; ob.y = w5 * inv; ob.z = w6 * inv; ob.w = w7 * inv;

    float* gdst = dst + (size_t)row * TIME + (size_t)tid * VPT;
    __builtin_nontemporal_store(oa, reinterpret_cast<f32x4*>(gdst));
    __builtin_nontemporal_store(ob, reinterpret_cast<f32x4*>(gdst + 4));
}

extern "C" void kernel_launch(void* const* d_in, const int* in_sizes, int n_in,
                              void* d_out, int out_size, void* d_ws, size_t ws_size,
                              hipStream_t stream) {
    const float* src = (const float*)d_in[0];
    float* dst = (float*)d_out;
    const int rows = in_sizes[0] / TIME;   // 65536 x 1024 per reference
    sparse_attn_topk_norm<<<dim3(rows), dim3(TPB), 0, stream>>>(src, dst, rows);
}